// TokenEncoder_12661563589328
// MI455X (gfx1250) — compile-verified
//
#include <hip/hip_runtime.h>
#include <hip/hip_bf16.h>

typedef float v2f __attribute__((ext_vector_type(2)));
typedef float v4f __attribute__((ext_vector_type(4)));
typedef float v8f __attribute__((ext_vector_type(8)));

#define D_MODEL 1024
#define S_SIG   16
#define N_TOK   4096
#define D_IN    256
#define B_SZ    16
#define L_SEQ   4096
#define LP1     (L_SEQ + 1)

#define M_BLK 128
#define N_BLK 128
#define K_BLK 32
#define LDA   36   // padded LDS row stride (floats): 16B-aligned rows, conflict-free frag reads

// ---------------------------------------------------------------------------
// Kernel 1: per-signal projection GEMM (fp32 WMMA 16x16x4) + bias + scatter
//   proj[s,n,e] = sum_d emb[s,n,d] * W[s,e,d] + b[s,e]
//   out[row(emb_index[s,n]), e] = proj[s,n,e]   with row(f) = (f/L)*(L+1) + f%L + 1
// ---------------------------------------------------------------------------
__global__ __launch_bounds__(256) void proj_scatter_kernel(
    const float* __restrict__ emb,        // (S, N, D_IN)
    const int*   __restrict__ emb_index,  // (S, N)
    const float* __restrict__ W,          // (S, D_MODEL, D_IN)
    const float* __restrict__ bias,       // (S, D_MODEL)
    float*       __restrict__ out)        // (B, L+1, D_MODEL)
{
    __shared__ float sA[M_BLK * LDA];
    __shared__ float sB[N_BLK * LDA];
    __shared__ int   sRow[M_BLK];
    __shared__ float sBias[N_BLK];

    const int s     = blockIdx.y;
    const int mtile = blockIdx.x & 31;   // 32 tiles over N_TOK
    const int ntile = blockIdx.x >> 5;   // 8 tiles over D_MODEL
    const int n0    = mtile * M_BLK;
    const int e0    = ntile * N_BLK;

    const int tid  = threadIdx.x;
    const int lane = tid & 31;
    const int wave = tid >> 5;
    const int wm   = (wave >> 2) * 64;   // wave M offset: 0 or 64
    const int wn   = (wave & 3) * 32;    // wave N offset: 0,32,64,96

    // Precompute scatter destination rows + bias slice (reconverges before WMMA)
    if (tid < M_BLK) {
        int f = emb_index[s * N_TOK + n0 + tid];
        int b = f / L_SEQ;
        int l = f - b * L_SEQ;
        sRow[tid] = b * LP1 + l + 1;
    } else {
        int i = tid - M_BLK;             // 0..127
        sBias[i] = bias[s * D_MODEL + e0 + i];
    }

    v8f acc[4][2];
#pragma unroll
    for (int mt = 0; mt < 4; ++mt)
#pragma unroll
        for (int nt = 0; nt < 2; ++nt)
            acc[mt][nt] = (v8f)0.0f;

    const float* __restrict__ Abase = emb + ((size_t)s * N_TOK   + n0) * D_IN;
    const float* __restrict__ Bbase = W   + ((size_t)s * D_MODEL + e0) * D_IN;

    // Fragment lane mapping for V_WMMA_F32_16X16X4_F32 (wave32):
    //   A (16x4): m = lane%16, k pair = 2*(lane/16)  -> {A[m][k], A[m][k+1]}
    //   B (4x16): n = lane%16, k pair = 2*(lane/16)  -> {B[k][n], B[k+1][n]}
    const int frag_mn = lane & 15;
    const int frag_k0 = (lane >> 4) << 1;

    for (int kc = 0; kc < D_IN; kc += K_BLK) {
        __syncthreads();
        // Stage 128x32 tiles of A (emb) and B (W, kept row-major [e][d]) into LDS.
        // 1024 float4s per tile, 256 threads -> 4 float4s each.
#pragma unroll
        for (int p = 0; p < 4; ++p) {
            int f   = tid + p * 256;      // 0..1023
            int row = f >> 3;             // 0..127
            int c   = (f & 7) << 2;       // 0,4,...,28
            v4f av = *(const v4f*)(Abase + (size_t)row * D_IN + kc + c);
            *(v4f*)(&sA[row * LDA + c]) = av;
            v4f bv = *(const v4f*)(Bbase + (size_t)row * D_IN + kc + c);
            *(v4f*)(&sB[row * LDA + c]) = bv;
        }
        __syncthreads();

#pragma unroll
        for (int ks = 0; ks < K_BLK; ks += 4) {
            v2f afr[4], bfr[2];
#pragma unroll
            for (int mt = 0; mt < 4; ++mt)
                afr[mt] = *(const v2f*)(&sA[(wm + mt * 16 + frag_mn) * LDA + ks + frag_k0]);
#pragma unroll
            for (int nt = 0; nt < 2; ++nt)
                bfr[nt] = *(const v2f*)(&sB[(wn + nt * 16 + frag_mn) * LDA + ks + frag_k0]);
#pragma unroll
            for (int mt = 0; mt < 4; ++mt)
#pragma unroll
                for (int nt = 0; nt < 2; ++nt)
                    acc[mt][nt] = __builtin_amdgcn_wmma_f32_16x16x4_f32(
                        /*neg_a=*/false, afr[mt], /*neg_b=*/false, bfr[nt],
                        /*c_mod=*/(short)0, acc[mt][nt],
                        /*reuse_a=*/false, /*reuse_b=*/false);
        }
    }

    // Epilogue: C/D layout -> row = v + 8*(lane/16), col = lane%16 per 16x16 tile.
    const int half = lane >> 4;
    const int coln = lane & 15;
#pragma unroll
    for (int mt = 0; mt < 4; ++mt) {
#pragma unroll
        for (int nt = 0; nt < 2; ++nt) {
#pragma unroll
            for (int v = 0; v < 8; ++v) {
                int rl   = wm + mt * 16 + v + 8 * half;   // local M row (0..127)
                int cl   = wn + nt * 16 + coln;           // local N col (0..127)
                int orow = sRow[rl];
                out[(size_t)orow * D_MODEL + e0 + cl] = acc[mt][nt][v] + sBias[cl];
            }
        }
    }
}

// ---------------------------------------------------------------------------
// Kernel 2: metadata embedding add + mask-zero + CLS row + attn_keep
// One 256-thread block per output row; each thread owns one float4 (1024/256).
// ---------------------------------------------------------------------------
__global__ __launch_bounds__(256) void meta_kernel(
    const int* __restrict__ pos, const int* __restrict__ sid,
    const int* __restrict__ mod, const int* __restrict__ role,
    const unsigned char* __restrict__ mask,
    const float* __restrict__ cls_content,
    const float* __restrict__ pos_table, const float* __restrict__ id_table,
    const float* __restrict__ mod_table, const float* __restrict__ role_table,
    float* __restrict__ out, float* __restrict__ attn)
{
    const int r   = blockIdx.x;          // 0 .. B*(L+1)-1
    const int b   = r / LP1;
    const int j   = r - b * LP1;
    const int tid = threadIdx.x;
    float* orow = out + (size_t)r * D_MODEL;
    const int i = tid * 4;

    if (j == 0) {                         // CLS row
        *(v4f*)(orow + i) = *(const v4f*)(cls_content + i);
        if (tid == 0) attn[r] = 1.0f;
        return;
    }

    const int l  = j - 1;
    const int ti = b * L_SEQ + l;
    __shared__ int sIdx[5];
    if (tid < 5) {
        switch (tid) {
            case 0: sIdx[0] = pos[ti];  break;
            case 1: sIdx[1] = sid[ti];  break;
            case 2: sIdx[2] = mod[ti];  break;
            case 3: sIdx[3] = role[ti]; break;
            case 4: sIdx[4] = (int)mask[ti]; break;
        }
    }
    __syncthreads();

    if (sIdx[4]) {                        // padded -> zero row
        v4f z = (v4f)0.0f;
        *(v4f*)(orow + i) = z;
        if (tid == 0) attn[r] = 0.0f;
        return;
    }

    v4f v = *(const v4f*)(orow + i);
    v += *(const v4f*)(pos_table  + (size_t)sIdx[0] * D_MODEL + i);
    v += *(const v4f*)(id_table   + (size_t)sIdx[1] * D_MODEL + i);
    v += *(const v4f*)(mod_table  + (size_t)sIdx[2] * D_MODEL + i);
    v += *(const v4f*)(role_table + (size_t)sIdx[3] * D_MODEL + i);
    *(v4f*)(orow + i) = v;
    if (tid == 0) attn[r] = 1.0f;
}

// ---------------------------------------------------------------------------
extern "C" void kernel_launch(void* const* d_in, const int* in_sizes, int n_in,
                              void* d_out, int out_size, void* d_ws, size_t ws_size,
                              hipStream_t stream) {
    (void)in_sizes; (void)n_in; (void)d_ws; (void)ws_size; (void)out_size;

    const float*         emb       = (const float*)d_in[0];
    const int*           emb_index = (const int*)d_in[1];
    const int*           pos       = (const int*)d_in[2];
    const int*           sid       = (const int*)d_in[3];
    const int*           mod       = (const int*)d_in[4];
    const int*           role      = (const int*)d_in[5];
    const unsigned char* mask      = (const unsigned char*)d_in[6];  // jnp bool = 1 byte
    const float*         W_proj    = (const float*)d_in[7];
    const float*         b_proj    = (const float*)d_in[8];
    const float*         cls       = (const float*)d_in[9];
    const float*         pos_t     = (const float*)d_in[10];
    const float*         id_t      = (const float*)d_in[11];
    const float*         mod_t     = (const float*)d_in[12];
    const float*         role_t    = (const float*)d_in[13];

    float* out  = (float*)d_out;                                  // (B, L+1, D_MODEL)
    float* attn = out + (size_t)B_SZ * LP1 * D_MODEL;             // (B, L+1) as float

    // GEMM+scatter: 32 M-tiles * 8 N-tiles per signal, 16 signals
    dim3 g1(256, S_SIG);
    proj_scatter_kernel<<<g1, 256, 0, stream>>>(emb, emb_index, W_proj, b_proj, out);

    // Metadata epilogue: one block per output row (same stream -> ordered)
    meta_kernel<<<B_SZ * LP1, 256, 0, stream>>>(pos, sid, mod, role, mask,
                                                cls, pos_t, id_t, mod_t, role_t,
                                                out, attn);
}